// SimplifiedDecoder_32461362823519
// MI455X (gfx1250) — compile-verified
//
#include <hip/hip_runtime.h>
#include <math.h>

typedef __attribute__((ext_vector_type(2))) float v2f;
typedef __attribute__((ext_vector_type(8))) float v8f;

#define B_ 256
#define H_ 128
#define V_ 16
#define N_ 200
#define W_ 8
#define NEGV (-1e9f)

__device__ __forceinline__ float sigm(float x) { return 1.f / (1.f + __expf(-x)); }

__device__ __forceinline__ v8f wmma_f32(v2f a, v2f b, v8f c) {
  // V_WMMA_F32_16X16X4_F32 : D = A(16x4) x B(4x16) + C(16x16), fp32 throughout
  return __builtin_amdgcn_wmma_f32_16x16x4_f32(false, a, false, b, (short)0, c, false, false);
}

// ---------------------------------------------------------------------------
// K1: veh_repr[b,v,h] = mean_n pairwise[b,h,v,n]   (streams the 400MB tensor)
// One wave per (b,h,v) row = exactly 100 aligned float2; non-temporal loads
// (row > L2-usefulness: 400MB stream vs 192MB L2) emit TH=NT global_load_b64.
// ---------------------------------------------------------------------------
__global__ __launch_bounds__(256) void k_reduce_mean(const float* __restrict__ P,
                                                     float* __restrict__ vr) {
  int wave = threadIdx.x >> 5, lane = threadIdx.x & 31;
  long row = (long)blockIdx.x * 8 + wave;            // row = (b*H + h)*V + v
  const v2f* src = (const v2f*)(P + row * N_);       // 800B row offset -> 8B aligned
  float s = 0.f;
  for (int n = lane; n < N_ / 2; n += 32) {          // 100 float2 per row
    v2f v = __builtin_nontemporal_load(src + n);
    s += v.x + v.y;
  }
  for (int off = 16; off; off >>= 1) s += __shfl_down(s, off, 32);
  if (lane == 0) {
    int v = (int)(row % V_);
    long bh = row / V_;
    int h = (int)(bh % H_);
    int b = (int)(bh / H_);
    vr[((long)b * V_ + v) * H_ + h] = s * (1.f / (float)N_);
  }
}

// ---------------------------------------------------------------------------
// Generic WMMA GEMM:  C[M,N] = act(A[M,K] @ W[N,K]^T + bias) (+ residual)
// One wave per 16x16 tile; K stepped by 4 with V_WMMA_F32_16X16X4_F32.
// act: 0=none, 1=tanh. residual added after activation.
// ---------------------------------------------------------------------------
__global__ __launch_bounds__(256) void k_wmma_gemm(
    const float* __restrict__ A, const float* __restrict__ W,
    const float* __restrict__ bias, const float* __restrict__ resid,
    float* __restrict__ C, int M, int N, int K, int act) {
  int wave = threadIdx.x >> 5, lane = threadIdx.x & 31;
  int l = lane & 15, half = lane >> 4;
  int tilesN = N >> 4;
  int tile = blockIdx.x * (blockDim.x >> 5) + wave;   // wave-uniform
  if (tile >= (M >> 4) * tilesN) return;              // whole wave exits: EXEC stays all-ones
  int m0 = (tile / tilesN) << 4;
  int n0 = (tile % tilesN) << 4;
  const float* ar = A + (long)(m0 + l) * K + 2 * half;
  const float* wr = W + (long)(n0 + l) * K + 2 * half;
  v8f acc = {};
  for (int k = 0; k < K; k += 4) {
    v2f a; a.x = ar[k]; a.y = ar[k + 1];
    v2f b; b.x = wr[k]; b.y = wr[k + 1];
    acc = wmma_f32(a, b, acc);
  }
  int col = n0 + l;
  float bv = bias ? bias[col] : 0.f;
#pragma unroll
  for (int j = 0; j < 8; ++j) {
    int r = m0 + j + 8 * half;
    float v = acc[j] + bv;
    if (act) v = tanhf(v);
    if (resid) v += resid[(long)r * N + col];
    C[(long)r * N + col] = v;
  }
}

// ---------------------------------------------------------------------------
// K2: fused LSTM cell: g = x@Wih^T + h0@Whh^T (WMMA) -> gates -> h_t,c_t,
//     plus h_mem_new = [h_mem[:,1:], h_t].  16 batch rows per block, 32 waves
//     cover the 16x512 gate tile.
// ---------------------------------------------------------------------------
__global__ __launch_bounds__(1024) void k_lstm(
    const float* __restrict__ x, const float* __restrict__ h0,
    const float* __restrict__ c0, const float* __restrict__ hmem_in,
    const float* __restrict__ w_ih, const float* __restrict__ w_hh,
    const float* __restrict__ b_ih, const float* __restrict__ b_hh,
    float* __restrict__ ht_ws, float* __restrict__ ct_ws, float* __restrict__ hmem_ws,
    float* __restrict__ ht_out, float* __restrict__ ct_out, float* __restrict__ hmem_out) {
  __shared__ float gbuf[16 * 512];
  __shared__ float hloc[16 * 128];
  int wave = threadIdx.x >> 5, lane = threadIdx.x & 31;
  int l = lane & 15, half = lane >> 4;
  int b0 = blockIdx.x * 16;
  int n0 = wave * 16;
  v8f acc = {};
  {
    const float* ar = x + (long)(b0 + l) * H_ + 2 * half;
    const float* wr = w_ih + (long)(n0 + l) * H_ + 2 * half;
    for (int k = 0; k < H_; k += 4) {
      v2f a; a.x = ar[k]; a.y = ar[k + 1];
      v2f b; b.x = wr[k]; b.y = wr[k + 1];
      acc = wmma_f32(a, b, acc);
    }
  }
  {
    const float* ar = h0 + (long)(b0 + l) * H_ + 2 * half;
    const float* wr = w_hh + (long)(n0 + l) * H_ + 2 * half;
    for (int k = 0; k < H_; k += 4) {
      v2f a; a.x = ar[k]; a.y = ar[k + 1];
      v2f b; b.x = wr[k]; b.y = wr[k + 1];
      acc = wmma_f32(a, b, acc);
    }
  }
#pragma unroll
  for (int j = 0; j < 8; ++j) gbuf[(j + 8 * half) * 512 + n0 + l] = acc[j];
  __syncthreads();
  for (int idx = threadIdx.x; idx < 16 * 128; idx += 1024) {
    int r = idx >> 7, hh = idx & 127;
    int b = b0 + r;
    float gi = gbuf[r * 512 + hh] + b_ih[hh] + b_hh[hh];
    float gf = gbuf[r * 512 + 128 + hh] + b_ih[128 + hh] + b_hh[128 + hh];
    float gg = gbuf[r * 512 + 256 + hh] + b_ih[256 + hh] + b_hh[256 + hh];
    float go = gbuf[r * 512 + 384 + hh] + b_ih[384 + hh] + b_hh[384 + hh];
    float c2 = sigm(gf) * c0[(long)b * H_ + hh] + sigm(gi) * tanhf(gg);
    float h = sigm(go) * tanhf(c2);
    hloc[idx] = h;
    ht_ws[(long)b * H_ + hh] = h;  ht_out[(long)b * H_ + hh] = h;
    ct_ws[(long)b * H_ + hh] = c2; ct_out[(long)b * H_ + hh] = c2;
  }
  __syncthreads();
  for (int idx = threadIdx.x; idx < 16 * W_ * H_; idx += 1024) {
    int r = idx >> 10;                  // W_*H_ = 1024
    int rem = idx & 1023;
    int t = rem >> 7, hh = rem & 127;
    int b = b0 + r;
    float v = (t < W_ - 1) ? hmem_in[((long)b * W_ + t + 1) * H_ + hh]
                           : hloc[r * 128 + hh];
    long o = ((long)b * W_ + t) * H_ + hh;
    hmem_ws[o] = v; hmem_out[o] = v;
  }
}

// ---------------------------------------------------------------------------
// K4: full 8-step GRU in one kernel (rows independent across batch).
//     gi_all (x@Wih^T + b_ih) precomputed; each step: gh = h@Whh^T + b_hh via
//     WMMA with A read from LDS-resident h; combine; accumulate mean.
// ---------------------------------------------------------------------------
__global__ __launch_bounds__(768) void k_gru(
    const float* __restrict__ gi_all, const float* __restrict__ w_hh,
    const float* __restrict__ b_hh, float* __restrict__ mem_mean) {
  __shared__ float hbuf[16 * 128];
  __shared__ float ghb[16 * 384];
  __shared__ float hsum[16 * 128];
  int wave = threadIdx.x >> 5, lane = threadIdx.x & 31;
  int l = lane & 15, half = lane >> 4;
  int b0 = blockIdx.x * 16;
  int n0 = wave * 16;
  for (int idx = threadIdx.x; idx < 16 * 128; idx += 768) { hbuf[idx] = 0.f; hsum[idx] = 0.f; }
  __syncthreads();
  for (int t = 0; t < W_; ++t) {
    v8f acc = {};
    const float* hr = hbuf + l * 128 + 2 * half;
    const float* wr = w_hh + (long)(n0 + l) * H_ + 2 * half;
    for (int k = 0; k < H_; k += 4) {
      v2f a; a.x = hr[k]; a.y = hr[k + 1];
      v2f b; b.x = wr[k]; b.y = wr[k + 1];
      acc = wmma_f32(a, b, acc);
    }
    int col = n0 + l;
    float bv = b_hh[col];
#pragma unroll
    for (int j = 0; j < 8; ++j) ghb[(j + 8 * half) * 384 + col] = acc[j] + bv;
    __syncthreads();
    for (int idx = threadIdx.x; idx < 16 * 128; idx += 768) {
      int r = idx >> 7, hh = idx & 127;
      int b = b0 + r;
      const float* gi = gi_all + ((long)b * W_ + t) * 384;
      float rg = sigm(gi[hh] + ghb[r * 384 + hh]);
      float zg = sigm(gi[128 + hh] + ghb[r * 384 + 128 + hh]);
      float nn = tanhf(gi[256 + hh] + rg * ghb[r * 384 + 256 + hh]);
      float h = hbuf[idx];
      float h2 = (1.f - zg) * nn + zg * h;
      hbuf[idx] = h2;
      hsum[idx] += h2;
    }
    __syncthreads();
  }
  for (int idx = threadIdx.x; idx < 16 * 128; idx += 768) {
    int r = idx >> 7, hh = idx & 127;
    mem_mean[(long)(b0 + r) * H_ + hh] = hsum[idx] * (1.f / (float)W_);
  }
}

// ---------------------------------------------------------------------------
// K8: vehicle attention (S=16, 4 heads, hd=32) per batch element.
//     Produces ctx (pre out-proj), veh_probs, veh_idx (argmax).
// ---------------------------------------------------------------------------
__global__ __launch_bounds__(128) void k_att_veh(
    const float* __restrict__ Q, const float* __restrict__ KV,
    const float* __restrict__ maskv, float* __restrict__ ctx,
    float* __restrict__ probs_out, float* __restrict__ vidx_out_f,
    int* __restrict__ vidx_ws) {
  int b = blockIdx.x, t = threadIdx.x;
  __shared__ float q[128];
  __shared__ float attn[64];
  __shared__ float vw[16];
  __shared__ float pr[16];
  q[t] = Q[(long)b * 128 + t];
  __syncthreads();
  if (t < 64) {
    int hd = t >> 4, s = t & 15;
    const float* kr = KV + ((long)b * 16 + s) * 256 + hd * 32;
    float sc = 0.f;
    for (int d = 0; d < 32; ++d) sc += q[hd * 32 + d] * kr[d];
    sc *= 0.17677669529663687f;                       // 1/sqrt(32)
    if (maskv[b * 16 + s] <= 0.5f) sc = NEGV;
    attn[t] = sc;
  }
  __syncthreads();
  if (t < 4) {                                        // softmax per head over 16 keys
    float m = -INFINITY;
    for (int s = 0; s < 16; ++s) m = fmaxf(m, attn[t * 16 + s]);
    float sm = 0.f;
    for (int s = 0; s < 16; ++s) { float e = __expf(attn[t * 16 + s] - m); attn[t * 16 + s] = e; sm += e; }
    float inv = 1.f / sm;
    for (int s = 0; s < 16; ++s) attn[t * 16 + s] *= inv;
  }
  __syncthreads();
  if (t < 16) vw[t] = 0.25f * (attn[t] + attn[16 + t] + attn[32 + t] + attn[48 + t]);
  __syncthreads();
  {                                                   // ctx[h] = sum_s attn * V
    int hd = t >> 5;
    const float* vbase = KV + (long)b * 16 * 256 + 128 + t;
    float s = 0.f;
    for (int k = 0; k < 16; ++k) s += attn[hd * 16 + k] * vbase[k * 256];
    ctx[(long)b * 128 + t] = s;
  }
  if (t == 0) {                                       // masked softmax of log(w+eps), argmax
    float lg[16]; float m = -INFINITY;
    for (int s = 0; s < 16; ++s) {
      lg[s] = (maskv[b * 16 + s] > 0.5f) ? __logf(vw[s] + 1e-8f) : NEGV;
      m = fmaxf(m, lg[s]);
    }
    float sm = 0.f;
    for (int s = 0; s < 16; ++s) { lg[s] = __expf(lg[s] - m); sm += lg[s]; }
    float inv = 1.f / sm; int best = 0; float bp = -1.f;
    for (int s = 0; s < 16; ++s) { float p = lg[s] * inv; pr[s] = p; if (p > bp) { bp = p; best = s; } }
    vidx_ws[b] = best;
    vidx_out_f[b] = (float)best;
  }
  __syncthreads();
  if (t < 16) probs_out[b * 16 + t] = pr[t];
}

// ---------------------------------------------------------------------------
// K10: LayerNorm over H=128 per row.
// ---------------------------------------------------------------------------
__global__ __launch_bounds__(128) void k_ln(const float* __restrict__ X,
                                            const float* __restrict__ g,
                                            const float* __restrict__ bb,
                                            float* __restrict__ Y) {
  __shared__ float red[128];
  int r = blockIdx.x, t = threadIdx.x;
  float v = X[(long)r * 128 + t];
  red[t] = v; __syncthreads();
  for (int s = 64; s; s >>= 1) { if (t < s) red[t] += red[t + s]; __syncthreads(); }
  float m = red[0] * (1.f / 128.f); __syncthreads();
  float d = v - m;
  red[t] = d * d; __syncthreads();
  for (int s = 64; s; s >>= 1) { if (t < s) red[t] += red[t + s]; __syncthreads(); }
  float var = red[0] * (1.f / 128.f);
  Y[(long)r * 128 + t] = d * rsqrtf(var + 1e-5f) * g[t] + bb[t];
}

// ---------------------------------------------------------------------------
// K12: fold W_k into the query:  qh[b,hd,:] = (1/sqrt32) * W_k_hd^T Q2_hd,
//      qc[b,hd] = (1/sqrt32) * Q2_hd . b_k_hd
// ---------------------------------------------------------------------------
__global__ __launch_bounds__(128) void k_qh(const float* __restrict__ Q2,
                                            const float* __restrict__ wk,
                                            const float* __restrict__ bk,
                                            float* __restrict__ qh,
                                            float* __restrict__ qc) {
  int bh = blockIdx.x; int b = bh >> 2, hd = bh & 3;
  __shared__ float qv[32];
  int t = threadIdx.x;
  if (t < 32) qv[t] = Q2[(long)b * 128 + hd * 32 + t];
  __syncthreads();
  float s = 0.f;
  for (int d = 0; d < 32; ++d) s += qv[d] * wk[(long)(hd * 32 + d) * 128 + t];
  qh[(long)bh * 128 + t] = s * 0.17677669529663687f;
  if (t < 32) {
    float p = qv[t] * bk[hd * 32 + t];
    for (int off = 16; off; off >>= 1) p += __shfl_down(p, off, 32);
    if (t == 0) qc[bh] = p * 0.17677669529663687f;
  }
}

// ---------------------------------------------------------------------------
// K13: customer attention probs. Streams the selected vehicle slice of
//      pairwise_embeds directly (no K matrix materialized).
// ---------------------------------------------------------------------------
__device__ __forceinline__ float blk_max(float v, float* red) {
  red[threadIdx.x] = v; __syncthreads();
  for (int s = 128; s; s >>= 1) { if (threadIdx.x < s) red[threadIdx.x] = fmaxf(red[threadIdx.x], red[threadIdx.x + s]); __syncthreads(); }
  float r = red[0]; __syncthreads(); return r;
}
__device__ __forceinline__ float blk_sum(float v, float* red) {
  red[threadIdx.x] = v; __syncthreads();
  for (int s = 128; s; s >>= 1) { if (threadIdx.x < s) red[threadIdx.x] += red[threadIdx.x + s]; __syncthreads(); }
  float r = red[0]; __syncthreads(); return r;
}

__global__ __launch_bounds__(256) void k_att_cust(
    const float* __restrict__ P, const float* __restrict__ maskp,
    const float* __restrict__ qh, const float* __restrict__ qc,
    const int* __restrict__ vidx, float* __restrict__ probs_out) {
  int b = blockIdx.x, t = threadIdx.x;
  __shared__ float qs[512];
  __shared__ float red[256];
  __shared__ int vs;
  if (t == 0) vs = vidx[b];
  for (int i = t; i < 512; i += 256) qs[i] = qh[(long)b * 512 + i];
  __syncthreads();
  int v = vs;
  float a0 = qc[b * 4 + 0], a1 = qc[b * 4 + 1], a2 = qc[b * 4 + 2], a3 = qc[b * 4 + 3];
  const float* p = P + ((long)b * (H_ * V_) + v) * N_;   // h=0 row of selected vehicle
  if (t < N_) {
    for (int h = 0; h < 128; ++h) {
      float pv = p[(long)h * (V_ * N_) + t];             // contiguous across threads
      a0 += qs[h] * pv;
      a1 += qs[128 + h] * pv;
      a2 += qs[256 + h] * pv;
      a3 += qs[384 + h] * pv;
    }
  }
  bool keep = (t < N_) && (maskp[((long)b * V_ + v) * N_ + t] > 0.5f);
  float sc[4] = {a0, a1, a2, a3};
  float w = 0.f;
  for (int hd = 0; hd < 4; ++hd) {
    float s = keep ? sc[hd] : NEGV;
    float m = blk_max((t < N_) ? s : NEGV, red);
    float e = (t < N_) ? __expf(s - m) : 0.f;
    float ssum = blk_sum(e, red);
    w += 0.25f * e / ssum;
  }
  float lg = keep ? __logf(w + 1e-8f) : NEGV;
  float m2 = blk_max((t < N_) ? lg : NEGV, red);
  float e2 = (t < N_) ? __expf(lg - m2) : 0.f;
  float s2 = blk_sum(e2, red);
  if (t < N_) probs_out[(long)b * N_ + t] = e2 / s2;
}

// ---------------------------------------------------------------------------
extern "C" void kernel_launch(void* const* d_in, const int* in_sizes, int n_in,
                              void* d_out, int out_size, void* d_ws, size_t ws_size,
                              hipStream_t stream) {
  (void)in_sizes; (void)n_in; (void)out_size; (void)ws_size;
  const float* P     = (const float*)d_in[0];
  const float* din   = (const float*)d_in[1];
  const float* h0    = (const float*)d_in[2];
  const float* c0    = (const float*)d_in[3];
  const float* hmem  = (const float*)d_in[4];
  const float* maskv = (const float*)d_in[5];
  const float* maskp = (const float*)d_in[6];
  const float* lw_ih = (const float*)d_in[7];
  const float* lw_hh = (const float*)d_in[8];
  const float* lb_ih = (const float*)d_in[9];
  const float* lb_hh = (const float*)d_in[10];
  const float* gw_ih = (const float*)d_in[11];
  const float* gw_hh = (const float*)d_in[12];
  const float* gb_ih = (const float*)d_in[13];
  const float* gb_hh = (const float*)d_in[14];
  const float* mem_w = (const float*)d_in[15];
  const float* mem_b = (const float*)d_in[16];
  const float* vin_w = (const float*)d_in[17];
  const float* vin_b = (const float*)d_in[18];
  const float* vout_w = (const float*)d_in[19];
  const float* vout_b = (const float*)d_in[20];
  const float* cin_w = (const float*)d_in[21];
  const float* cin_b = (const float*)d_in[22];
  const float* lnv_g = (const float*)d_in[25];
  const float* lnv_b = (const float*)d_in[26];
  // d_in[23],[24],[27],[28] (cust out-proj / ln_cust) feed cust_out, unused in returns.

  float* out = (float*)d_out;
  float* ws  = (float*)d_ws;
  float* VR   = ws;                 // [B,V,H]      524288
  float* HT   = VR + 524288;        // [B,H]        32768
  float* CT   = HT + 32768;         // [B,H]        32768
  float* HM   = CT + 32768;         // [B,W,H]      262144
  float* GI   = HM + 262144;        // [B*W,3H]     786432
  float* MM   = GI + 786432;        // [B,H]        32768
  float* HE   = MM + 32768;         // [B,H]        32768
  float* Qv   = HE + 32768;         // [B,H]        32768
  float* KV   = Qv + 32768;         // [B*V,2H]     1048576
  float* CTX  = KV + 1048576;       // [B,H]        32768
  float* PROJ = CTX + 32768;        // [B,H]        32768
  float* VO   = PROJ + 32768;       // [B,H]        32768
  float* Q2   = VO + 32768;         // [B,H]        32768
  float* QH   = Q2 + 32768;         // [B,4,H]      131072
  float* QC   = QH + 131072;        // [B,4]        1024
  int*   VIDX = (int*)(QC + 1024);  // [B]

  float* out_vprob = out;               // [B,V]   4096
  float* out_cprob = out + 4096;        // [B,N]   51200
  float* out_vidx  = out + 55296;       // [B]     256
  float* out_ht    = out + 55552;       // [B,H]
  float* out_ct    = out + 88320;       // [B,H]
  float* out_hm    = out + 121088;      // [B,W,H]

  // Phase 1 (independent streams): big reduction + recurrent chain
  k_reduce_mean<<<65536, 256, 0, stream>>>(P, VR);
  k_lstm<<<16, 1024, 0, stream>>>(din, h0, c0, hmem, lw_ih, lw_hh, lb_ih, lb_hh,
                                  HT, CT, HM, out_ht, out_ct, out_hm);
  k_wmma_gemm<<<384, 256, 0, stream>>>(HM, gw_ih, gb_ih, nullptr, GI, 2048, 384, 128, 0);
  k_gru<<<16, 768, 0, stream>>>(GI, gw_hh, gb_hh, MM);
  k_wmma_gemm<<<16, 256, 0, stream>>>(MM, mem_w, mem_b, HT, HE, 256, 128, 128, 1); // h_enriched
  // Phase 2: vehicle attention
  k_wmma_gemm<<<16, 256, 0, stream>>>(HE, vin_w, vin_b, nullptr, Qv, 256, 128, 128, 0);
  k_wmma_gemm<<<512, 256, 0, stream>>>(VR, vin_w + 128 * 128, vin_b + 128, nullptr, KV,
                                       4096, 256, 128, 0);
  k_att_veh<<<256, 128, 0, stream>>>(Qv, KV, maskv, CTX, out_vprob, out_vidx, VIDX);
  k_wmma_gemm<<<16, 256, 0, stream>>>(CTX, vout_w, vout_b, nullptr, PROJ, 256, 128, 128, 0);
  k_ln<<<256, 128, 0, stream>>>(PROJ, lnv_g, lnv_b, VO);
  // Phase 3: customer attention (K-projection folded into query)
  k_wmma_gemm<<<16, 256, 0, stream>>>(VO, cin_w, cin_b, nullptr, Q2, 256, 128, 128, 0);
  k_qh<<<1024, 128, 0, stream>>>(Q2, cin_w + 128 * 128, cin_b + 128, QH, QC);
  k_att_cust<<<256, 256, 0, stream>>>(P, maskp, QH, QC, VIDX, out_cprob);
}